// SocialGAT_45226005627087
// MI455X (gfx1250) — compile-verified
//
#include <hip/hip_runtime.h>

typedef __attribute__((ext_vector_type(2))) float v2f;
typedef __attribute__((ext_vector_type(8))) float v8f;
typedef __attribute__((ext_vector_type(4))) int   v4i;

#define BATCH   8
#define AGENTS  50
#define TSTEPS  256
#define DIM     128
#define HEADS   32
#define NEG_SLOPE 0.2f
#define ROWS    (BATCH * AGENTS * TSTEPS)   /* 102400 GEMM rows */

// ---------------- async global -> LDS helpers (CDNA5 ASYNCcnt path) --------
__device__ __forceinline__ void async_g2l_b128(void* lds, const void* gsrc) {
#if __has_builtin(__builtin_amdgcn_global_load_async_to_lds_b128)
  __builtin_amdgcn_global_load_async_to_lds_b128(
      (__attribute__((address_space(1))) v4i*)gsrc,
      (__attribute__((address_space(3))) v4i*)lds, 0, 0);
#else
  *(float4*)lds = *(const float4*)gsrc;
#endif
}

__device__ __forceinline__ void async_wait_all() {
#if __has_builtin(__builtin_amdgcn_s_wait_asynccnt)
  __builtin_amdgcn_s_wait_asynccnt(0);
#elif __has_builtin(__builtin_amdgcn_global_load_async_to_lds_b128)
  asm volatile("s_wait_asynccnt 0" ::: "memory");
#endif
}

// ---------------------------------------------------------------------------
// Kernel 1: x = h @ W^T  (M=102400, K=128, N=128), exact fp32 via
// V_WMMA_F32_16X16X4_F32.  One block = 128 rows x 128 cols; wave w owns the
// 16-row sub-tile w, accumulating 8 N-tiles (8 x v8f accumulators).
// K is staged through LDS in 32-wide slabs with async DMA.
// ---------------------------------------------------------------------------
#define KCH  32
#define LSTR 36   /* padded LDS row stride (floats): 144 B, 16B aligned, conflict-free */

__global__ __launch_bounds__(256) void gat_lin_wmma(
    const float* __restrict__ h, const float* __restrict__ W,
    float* __restrict__ x)
{
  __shared__ float sA[128 * LSTR];
  __shared__ float sB[128 * LSTR];

  const int tid  = threadIdx.x;
  const int wave = tid >> 5;           // 0..7 -> M sub-tile
  const int lane = tid & 31;
  const int m    = lane & 15;          // A/B: row-in-tile / col-in-tile
  const int kh   = lane >> 4;          // K half select (K = v + 2*kh)
  const int r0   = blockIdx.x * 128;
  const float* __restrict__ hg = h + (size_t)r0 * DIM;

  v8f acc[8] = {};

  for (int kc = 0; kc < DIM; kc += KCH) {
    // ---- stage K-slab of the 128-row h tile and of W into LDS (async) ----
#pragma unroll
    for (int i = tid; i < 128 * (KCH / 4); i += 256) {
      int row = i >> 3;
      int c4  = (i & 7) << 2;
      async_g2l_b128(&sA[row * LSTR + c4], hg + row * DIM + kc + c4);
      async_g2l_b128(&sB[row * LSTR + c4], W  + row * DIM + kc + c4);
    }
    async_wait_all();
    __syncthreads();

    // ---- 8 K-steps of fp32 WMMA over all 8 N-tiles ----
#pragma unroll
    for (int k = 0; k < KCH; k += 4) {
      // A 16x4: lane(m, kh) holds K = k + 2*kh + {0,1}
      v2f a = *(const v2f*)&sA[(wave * 16 + m) * LSTR + k + 2 * kh];
#pragma unroll
      for (int nt = 0; nt < 8; ++nt) {
        // B 4x16: B[K][n] = W[n][K]; lane(m=n, kh) holds K = k + 2*kh + {0,1}
        v2f b = *(const v2f*)&sB[(nt * 16 + m) * LSTR + k + 2 * kh];
        acc[nt] = __builtin_amdgcn_wmma_f32_16x16x4_f32(
            false, a, false, b, (short)0, acc[nt], false, false);
      }
    }
    __syncthreads();
  }

  // ---- D writeout: C/D layout -> vgpr v holds row v + 8*kh, col = m ----
  const int mrow = r0 + wave * 16 + 8 * kh;
#pragma unroll
  for (int nt = 0; nt < 8; ++nt) {
    int ncol = nt * 16 + m;
#pragma unroll
    for (int v = 0; v < 8; ++v)
      x[(size_t)(mrow + v) * DIM + ncol] = acc[nt][v];
  }
}

// ---------------------------------------------------------------------------
// Kernel 2: per-graph GAT attention.  One block per graph (b,t); the 50x128
// transformed features are async-staged into LDS, logits a_src/a_dst are
// computed per (node, head) in LDS, then each wave owns a destination node i
// with lane == head: masked leaky-ReLU softmax over sources j != i and the
// weighted aggregation (C=4 floats per head per lane), bias added on store.
// ---------------------------------------------------------------------------
__global__ __launch_bounds__(256) void gat_attn(
    const float* __restrict__ x, const float* __restrict__ att_src,
    const float* __restrict__ att_dst, const float* __restrict__ bias,
    float* __restrict__ out)
{
  __shared__ float xs[AGENTS * DIM];
  __shared__ float asrc[AGENTS * HEADS];
  __shared__ float adst[AGENTS * HEADS];
  __shared__ float satts[DIM], sattd[DIM], sbias[DIM];

  const int tid = threadIdx.x;
  const int n   = blockIdx.x;          // graph id = b*T + t
  const int b   = n / TSTEPS;
  const int t   = n % TSTEPS;

  // stage this graph's rows x[(b*A + j)*T + t, :]  (512 B per row, coalesced)
  for (int i = tid; i < AGENTS * (DIM / 4); i += 256) {
    int j  = i >> 5;
    int c4 = (i & 31) << 2;
    const float* g = x + ((size_t)(b * AGENTS + j) * TSTEPS + t) * DIM + c4;
    async_g2l_b128(&xs[j * DIM + c4], g);
  }
  if (tid < DIM) {
    satts[tid] = att_src[tid];
    sattd[tid] = att_dst[tid];
    sbias[tid] = bias[tid];
  }
  async_wait_all();
  __syncthreads();

  // attention logits: a_src/a_dst[j,h] = <x[j,h,:], att[h,:]>
  for (int i = tid; i < AGENTS * HEADS; i += 256) {
    int j = i >> 5, hh = i & 31;
    const float4 xv = *(const float4*)&xs[j * DIM + hh * 4];
    asrc[i] = xv.x * satts[hh*4+0] + xv.y * satts[hh*4+1]
            + xv.z * satts[hh*4+2] + xv.w * satts[hh*4+3];
    adst[i] = xv.x * sattd[hh*4+0] + xv.y * sattd[hh*4+1]
            + xv.z * sattd[hh*4+2] + xv.w * sattd[hh*4+3];
  }
  __syncthreads();

  const int wave = tid >> 5;
  const int hh   = tid & 31;           // lane == head
  for (int i = wave; i < AGENTS; i += 8) {   // wave-uniform: EXEC stays full
    const float ad = adst[i * HEADS + hh];

    float mx = -3.0e38f;
    for (int j = 0; j < AGENTS; ++j) {
      if (j == i) continue;
      float e = ad + asrc[j * HEADS + hh];
      e = (e > 0.0f) ? e : NEG_SLOPE * e;
      mx = fmaxf(mx, e);
    }

    float s = 0.0f;
    float4 accv = make_float4(0.0f, 0.0f, 0.0f, 0.0f);
    for (int j = 0; j < AGENTS; ++j) {
      if (j == i) continue;
      float e = ad + asrc[j * HEADS + hh];
      e = (e > 0.0f) ? e : NEG_SLOPE * e;
      float w = __expf(e - mx);
      s += w;
      const float4 xv = *(const float4*)&xs[j * DIM + hh * 4];
      accv.x += w * xv.x; accv.y += w * xv.y;
      accv.z += w * xv.z; accv.w += w * xv.w;
    }

    const float inv = 1.0f / s;
    float4 o;
    o.x = accv.x * inv + sbias[hh*4+0];
    o.y = accv.y * inv + sbias[hh*4+1];
    o.z = accv.z * inv + sbias[hh*4+2];
    o.w = accv.w * inv + sbias[hh*4+3];
    *(float4*)&out[((size_t)(b * AGENTS + i) * TSTEPS + t) * DIM + hh * 4] = o;
  }
}

// ---------------------------------------------------------------------------
extern "C" void kernel_launch(void* const* d_in, const int* in_sizes, int n_in,
                              void* d_out, int out_size, void* d_ws, size_t ws_size,
                              hipStream_t stream) {
  const float* h       = (const float*)d_in[0];   // [B, A, T, D]
  const float* W       = (const float*)d_in[1];   // [H*C, D]
  const float* att_src = (const float*)d_in[2];   // [H, C]
  const float* att_dst = (const float*)d_in[3];   // [H, C]
  const float* bias    = (const float*)d_in[4];   // [H*C]
  float* out  = (float*)d_out;                    // [B, A, T, D]
  float* xbuf = (float*)d_ws;                     // [ROWS, 128] = 52.4 MB scratch

  gat_lin_wmma<<<ROWS / 128, 256, 0, stream>>>(h, W, xbuf);
  gat_attn<<<BATCH * TSTEPS, 256, 0, stream>>>(xbuf, att_src, att_dst, bias, out);
}